// TopRankRegularization_21809843929138
// MI455X (gfx1250) — compile-verified
//
#include <hip/hip_runtime.h>
#include <cstdint>
#include <cstddef>

// ---------------- tuning ----------------
#define STAGE_ROWS 256     // rows per pipeline stage (== blockDim)
#define STAGES     4       // stages per block -> 1024 rows/block in kA
#define ROW_F      20      // padded floats per row in LDS (80 B, 16B-aligned, kills 4-way conflicts)
#define ROW_B      80

__device__ __forceinline__ unsigned lds_byte_off(const void* p) {
  // Flat LDS aperture keeps the wave-relative LDS offset in the low 32 bits.
  return (unsigned)(size_t)p;
}

// ---------------------------------------------------------------------------
// Kernel A: streaming softmax-top1 pass.
// Double-buffered GLOBAL_LOAD_ASYNC_TO_LDS_B128 pipeline (CDNA5 async path),
// non-temporal (single-use 128 MB stream -> keep L2 for pv/cp/cs reuse).
// Writes pv[row] = +p if row selected (argmax==class0) else -p,
// and per-256-row-chunk selected counts.
// ---------------------------------------------------------------------------
__global__ __launch_bounds__(256) void kA_rowpass(const float* __restrict__ x,
                                                  float* __restrict__ pv,
                                                  unsigned* __restrict__ cnt,
                                                  int nrows)
{
  __shared__ __align__(16) float buf[2][STAGE_ROWS * ROW_F];
  __shared__ unsigned wsum[8];

  const int t = threadIdx.x;
  const unsigned lane = (unsigned)t & 31u;
  const unsigned wid  = (unsigned)t >> 5;
  const int base_row  = blockIdx.x * (STAGE_ROWS * STAGES);
  const unsigned lbase = lds_byte_off(&buf[0][0]);

  auto issue = [&](int s) {
    int r  = base_row + s * STAGE_ROWS + t;
    int rc = (r < nrows) ? r : (nrows - 1);          // clamp: keep EXEC full for async ops
    unsigned goff = (unsigned)rc * 64u;              // 16 floats * 4B
    unsigned loff = lbase + (unsigned)(s & 1) * (STAGE_ROWS * ROW_B) + (unsigned)t * ROW_B;
#pragma unroll
    for (int k = 0; k < 4; ++k) {
      asm volatile("global_load_async_to_lds_b128 %0, %1, %2 th:TH_LOAD_NT"
                   :: "v"(loff + 16u * (unsigned)k),
                      "v"(goff + 16u * (unsigned)k),
                      "s"(x)
                   : "memory");
    }
  };

  issue(0);
#pragma unroll
  for (int s = 0; s < STAGES; ++s) {
    if (s + 1 < STAGES) {
      issue(s + 1);
      asm volatile("s_wait_asynccnt 0x4" ::: "memory");   // stage s landed, s+1 in flight
    } else {
      asm volatile("s_wait_asynccnt 0x0" ::: "memory");
    }

    const float4* rp = (const float4*)&buf[s & 1][t * ROW_F];
    float4 a = rp[0], b = rp[1], c = rp[2], d = rp[3];

    float m = fmaxf(fmaxf(fmaxf(fmaxf(a.x, a.y), fmaxf(a.z, a.w)),
                          fmaxf(fmaxf(b.x, b.y), fmaxf(b.z, b.w))),
                    fmaxf(fmaxf(fmaxf(c.x, c.y), fmaxf(c.z, c.w)),
                          fmaxf(fmaxf(d.x, d.y), fmaxf(d.z, d.w))));

    float sum = __expf(a.x - m) + __expf(a.y - m) + __expf(a.z - m) + __expf(a.w - m)
              + __expf(b.x - m) + __expf(b.y - m) + __expf(b.z - m) + __expf(b.w - m)
              + __expf(c.x - m) + __expf(c.y - m) + __expf(c.z - m) + __expf(c.w - m)
              + __expf(d.x - m) + __expf(d.y - m) + __expf(d.z - m) + __expf(d.w - m);

    float p = 1.0f / sum;                      // top-1 softmax prob (sum in [1,16])
    int  r  = base_row + s * STAGE_ROWS + t;
    bool sel = (r < nrows) && (a.x >= m);      // class0 is (first) argmax
    if (r < nrows) pv[r] = sel ? p : -p;       // sign encodes the flag; p > 0 always

    unsigned bal = __builtin_amdgcn_ballot_w32(sel);
    if (lane == 0) wsum[wid] = (unsigned)__popc(bal);
    __syncthreads();
    if (t == 0) {
      unsigned tot = 0;
#pragma unroll
      for (int w = 0; w < 8; ++w) tot += wsum[w];
      cnt[blockIdx.x * STAGES + s] = tot;      // one count per 256-row chunk
    }
    __syncthreads();
  }
}

// ---------------------------------------------------------------------------
// Kernel B: single-block exclusive scan of chunk counts; offs[n] = total.
// Slice-serial + one 1024-wide scan: ~20 barriers total (vs ~160 before).
// ---------------------------------------------------------------------------
__global__ __launch_bounds__(1024) void kB_scan(const unsigned* __restrict__ cnt,
                                                unsigned* __restrict__ offs, int n)
{
  __shared__ unsigned tsum[1024];
  const int t   = threadIdx.x;
  const int per = (n + 1023) >> 10;            // contiguous elements per thread
  const int beg = t * per;

  unsigned s = 0;
  for (int k = 0; k < per; ++k) {
    int i = beg + k;
    if (i < n) s += cnt[i];
  }
  tsum[t] = s;
  __syncthreads();
  for (int d = 1; d < 1024; d <<= 1) {         // Hillis-Steele inclusive scan
    unsigned u = (t >= d) ? tsum[t - d] : 0u;
    __syncthreads();
    tsum[t] += u;
    __syncthreads();
  }
  unsigned run = tsum[t] - s;                  // exclusive base for this slice
  for (int k = 0; k < per; ++k) {
    int i = beg + k;
    if (i < n) { offs[i] = run; run += cnt[i]; }
  }
  if (t == 1023) offs[n] = tsum[1023];         // total selected
}

// ---------------------------------------------------------------------------
// Kernel C: order-preserving scatter of selected (p, score) into compact arrays.
// One block == one 256-row chunk; ballot-based intra-block ranks.
// ---------------------------------------------------------------------------
__global__ __launch_bounds__(256) void kC_scatter(const float* __restrict__ pv,
                                                  const float* __restrict__ score,
                                                  const unsigned* __restrict__ offs,
                                                  float* __restrict__ cp,
                                                  float* __restrict__ cs,
                                                  int nrows)
{
  __shared__ unsigned wsum[8];
  __shared__ unsigned woff[8];
  const int t = threadIdx.x;
  const unsigned lane = (unsigned)t & 31u;
  const unsigned wid  = (unsigned)t >> 5;
  int r = blockIdx.x * 256 + t;
  float v = (r < nrows) ? pv[r] : -1.0f;
  bool sel = v > 0.0f;
  unsigned bal = __builtin_amdgcn_ballot_w32(sel);
  if (lane == 0) wsum[wid] = (unsigned)__popc(bal);
  __syncthreads();
  if (t == 0) {
    unsigned run = 0;
#pragma unroll
    for (int w = 0; w < 8; ++w) { woff[w] = run; run += wsum[w]; }
  }
  __syncthreads();
  if (sel) {
    unsigned pos = offs[blockIdx.x] + woff[wid]
                 + (unsigned)__popc(bal & ((1u << lane) - 1u));
    cp[pos] = v;
    cs[pos] = score[r];
  }
}

// ---------------------------------------------------------------------------
// Kernel D: adjacent-pair ranking loss over the selected prefix; block partials.
// ---------------------------------------------------------------------------
__global__ __launch_bounds__(256) void kD_pairs(const float* __restrict__ cp,
                                                const float* __restrict__ cs,
                                                const unsigned* __restrict__ nsel_p,
                                                float* __restrict__ partial,
                                                int npairs_max)
{
  __shared__ float red[256];
  const int t = threadIdx.x;
  int i = blockIdx.x * 256 + t;
  unsigned nsel = *nsel_p;
  float loss = 0.0f;
  if (i < npairs_max && (unsigned)(i + 1) < nsel) {
    float p0 = cp[i], p1 = cp[i + 1];
    float s0 = cs[i], s1 = cs[i + 1];
    float dsv = s0 - s1;
    float sg = (dsv > 0.0f) ? 1.0f : ((dsv < 0.0f) ? -1.0f : 0.0f);
    float tm = (p0 - p1) * sg;                 // |tm| <= 1, no overflow
    loss = log1pf(__expf(-tm));
  }
  red[t] = loss;
  __syncthreads();
#pragma unroll
  for (int d = 128; d > 0; d >>= 1) {
    if (t < d) red[t] += red[t + d];
    __syncthreads();
  }
  if (t == 0) partial[blockIdx.x] = red[0];
}

// ---------------------------------------------------------------------------
// Kernel E: deterministic final reduction of block partials.
// ---------------------------------------------------------------------------
__global__ __launch_bounds__(1024) void kE_final(const float* __restrict__ partial,
                                                 float* __restrict__ out, int n)
{
  __shared__ float red[1024];
  const int t = threadIdx.x;
  float s = 0.0f;
  for (int i = t; i < n; i += 1024) s += partial[i];
  red[t] = s;
  __syncthreads();
#pragma unroll
  for (int d = 512; d > 0; d >>= 1) {
    if (t < d) red[t] += red[t + d];
    __syncthreads();
  }
  if (t == 0) out[0] = red[0];
}

// ---------------------------------------------------------------------------
extern "C" void kernel_launch(void* const* d_in, const int* in_sizes, int n_in,
                              void* d_out, int out_size, void* d_ws, size_t ws_size,
                              hipStream_t stream)
{
  const float* x     = (const float*)d_in[0];   // [B,16] f32 logits
  const float* score = (const float*)d_in[1];   // [B]    f32
  const int B = in_sizes[1];

  const int nblkA  = (B + (STAGE_ROWS * STAGES) - 1) / (STAGE_ROWS * STAGES);
  const int nchunk = nblkA * STAGES;            // 256-row chunks
  const int npairs = B - 1;
  const int nblkD  = (npairs + 255) / 256;

  char* w = (char*)d_ws;
  float*    pv      = (float*)w;     w += (size_t)B * 4;
  float*    cp      = (float*)w;     w += (size_t)B * 4;
  float*    cs      = (float*)w;     w += (size_t)B * 4;
  unsigned* cnt     = (unsigned*)w;  w += (size_t)nchunk * 4;
  unsigned* offs    = (unsigned*)w;  w += (size_t)(nchunk + 1) * 4;
  float*    partial = (float*)w;     w += (size_t)nblkD * 4;

  kA_rowpass<<<nblkA, 256, 0, stream>>>(x, pv, cnt, B);
  kB_scan<<<1, 1024, 0, stream>>>(cnt, offs, nchunk);
  kC_scatter<<<nchunk, 256, 0, stream>>>(pv, score, offs, cp, cs, B);
  kD_pairs<<<nblkD, 256, 0, stream>>>(cp, cs, offs + nchunk, partial, npairs);
  kE_final<<<1, 1024, 0, stream>>>(partial, (float*)d_out, nblkD);
}